// LossSSD_83889301226086
// MI455X (gfx1250) — compile-verified
//
#include <hip/hip_runtime.h>
#include <hip/hip_bf16.h>
#include <math.h>

#define BATCH 128
#define HW    8732
#define NCLS  21
#define NGT   16
#define TPB   256
#define NBLK  ((HW + TPB - 1) / TPB)   // 35
#define EPS16F 0.0009765625f

typedef __attribute__((ext_vector_type(2))) float v2f;
typedef __attribute__((ext_vector_type(8))) float v8f;

// ---- wave32 sum via V_WMMA_F32_16X16X4_F32 ----
// A (16x4): VGPR0 lanes0-15 -> A[m][0]=v(m), lanes16-31 -> A[m][2]=v(m+16); VGPR1 = 0.
// B (4x16) = all ones  =>  D[m][n] = v(m) + v(m+16).
// D layout: VGPR j: lanes0-15 hold M=j, lanes16-31 hold M=j+8, so summing the 8 D
// registers per lane gives half-sums; one shfl_xor(16) completes the reduction.
__device__ __forceinline__ float wave_sum_wmma(float v) {
    v2f a;  a.x = v;    a.y = 0.0f;
    v2f bm; bm.x = 1.0f; bm.y = 1.0f;
    v8f c = {};
    v8f d = __builtin_amdgcn_wmma_f32_16x16x4_f32(false, a, false, bm, (short)0, c, false, false);
    float t = ((d[0] + d[1]) + (d[2] + d[3])) + ((d[4] + d[5]) + (d[6] + d[7]));
    t += __shfl_xor(t, 16, 32);
    return t;
}

__global__ void k_init(unsigned long long* best) {
    int i = blockIdx.x * blockDim.x + threadIdx.x;
    if (i < BATCH * NGT) best[i] = 0ull;
}

// IoU of every anchor vs 16 GT boxes; per-anchor max/argmax; per-GT best anchor
// via u64 atomicMax with key = (iou_bits<<32) | (0xFFFFFFFF - hw)  (lowest hw wins ties).
__global__ void k_iou(const float* __restrict__ gboxes, const float* __restrict__ ancs,
                      float* __restrict__ max_iou, int* __restrict__ bidx,
                      unsigned long long* __restrict__ best) {
    int b = blockIdx.y, tid = threadIdx.x;
    int hw = blockIdx.x * TPB + tid;
    __shared__ float sg[NGT * 4];
    __shared__ unsigned long long sbest[NGT];
    if (tid < NGT) sbest[tid] = 0ull;
    if (tid < NGT * 4) sg[tid] = gboxes[b * NGT * 4 + tid];
    __syncthreads();
    if (hw < HW) {
        float ax = ancs[hw*4+0], ay = ancs[hw*4+1], aw = ancs[hw*4+2], ah = ancs[hw*4+3];
        float al = ax - aw*0.5f, at = ay - ah*0.5f, ar = ax + aw*0.5f, ab = ay + ah*0.5f;
        float area_a = aw * ah;
        float bi = -1.0f; int bg = 0;
        #pragma unroll
        for (int g = 0; g < NGT; ++g) {
            float gl = sg[g*4+0], gt = sg[g*4+1], gr = sg[g*4+2], gb = sg[g*4+3];
            float il = fmaxf(al, gl), it = fmaxf(at, gt);
            float ir = fminf(ar, gr), ib = fminf(ab, gb);
            float iw = fmaxf(ir - il, 0.0f), ih = fmaxf(ib - it, 0.0f);
            float inter  = iw * ih;
            float area_g = (gr - gl) * (gb - gt);
            float iou = inter / (area_a + area_g - inter + 1e-8f);
            if (iou > bi) { bi = iou; bg = g; }   // first-wins on ties (argmax semantics)
            unsigned long long key = ((unsigned long long)__float_as_uint(iou) << 32)
                                   | (unsigned long long)(0xFFFFFFFFu - (unsigned)hw);
            atomicMax(&sbest[g], key);
        }
        max_iou[(size_t)b * HW + hw] = bi;
        bidx[(size_t)b * HW + hw]    = bg;
    }
    __syncthreads();
    if (tid < NGT) atomicMax(&best[b * NGT + tid], sbest[tid]);
}

// Force each GT's best anchor positive; serial per-batch loop => deterministic last-wins.
__global__ void k_force(const unsigned long long* __restrict__ best,
                        float* __restrict__ max_iou, int* __restrict__ bidx) {
    int b = blockIdx.x * blockDim.x + threadIdx.x;
    if (b >= BATCH) return;
    for (int g = 0; g < NGT; ++g) {
        unsigned long long key = best[b * NGT + g];
        unsigned hw = 0xFFFFFFFFu - (unsigned)(key & 0xFFFFFFFFull);
        if (hw < (unsigned)HW) {
            max_iou[(size_t)b * HW + hw] = 2.0f;   // > IOU_THRESH => positive
            bidx[(size_t)b * HW + hw]    = g;
        }
    }
}

// Streaming pass: logsumexp CE, smooth-L1 on positives, neg-mining scores.
// Block partials reduced with the WMMA wave reduction, written per (batch, block).
__global__ void __launch_bounds__(TPB)
k_main(const float* __restrict__ preg, const float* __restrict__ pcls,
       const float* __restrict__ gboxes, const int* __restrict__ glabels,
       const float* __restrict__ ancs, const float* __restrict__ max_iou,
       const int* __restrict__ bidx, float* __restrict__ neg, float* __restrict__ part) {
    int b = blockIdx.y, tid = threadIdx.x;
    int hw = blockIdx.x * TPB + tid;
    float p_sl1 = 0.0f, p_ce = 0.0f, p_np = 0.0f;
    if (hw < HW) {
        size_t base = (size_t)b * HW + hw;
        float mi = max_iou[base];
        bool pos = mi > 0.5f;
        int gi  = bidx[base];
        int lab = pos ? glabels[b * NGT + gi] : 0;
        const float* pc = pcls + (size_t)b * NCLS * HW + hw;
        __builtin_prefetch(pc, 0, 1);                   // global_prefetch_b8
        float vals[NCLS];
        float vmax = -3.402823466e38f;
        #pragma unroll
        for (int c = 0; c < NCLS; ++c) { vals[c] = pc[(size_t)c * HW]; vmax = fmaxf(vmax, vals[c]); }
        float s = 0.0f;
        #pragma unroll
        for (int c = 0; c < NCLS; ++c) s += expf(vals[c] - vmax);
        float ce = logf(s) + vmax - vals[lab];
        neg[base] = pos ? -1e9f : ce;
        if (pos) {
            p_np = 1.0f; p_ce = ce;
            const float* gb = gboxes + (size_t)(b * NGT + gi) * 4;
            float gl = gb[0], gt = gb[1], gr = gb[2], gbt = gb[3];
            float gx = (gl + gr) * 0.5f, gy = (gt + gbt) * 0.5f;
            float gw = gr - gl,          gh = gbt - gt;
            float ax = ancs[hw*4+0], ay = ancs[hw*4+1], aw = ancs[hw*4+2], ah = ancs[hw*4+3];
            float tx = (gx - ax) / (0.1f * aw);
            float ty = (gy - ay) / (0.1f * ah);
            float tw = logf(gw / aw + 1e-8f) / 0.2f;
            float th = logf(gh / ah + 1e-8f) / 0.2f;
            const float* pr = preg + (size_t)b * 4 * HW + hw;
            float d0 = pr[0]              - tx;
            float d1 = pr[(size_t)HW]     - ty;
            float d2 = pr[(size_t)2 * HW] - tw;
            float d3 = pr[(size_t)3 * HW] - th;
            float sl1 = 0.0f, ad;
            ad = fabsf(d0); sl1 += (ad < 1.0f) ? 0.5f * d0 * d0 : ad - 0.5f;
            ad = fabsf(d1); sl1 += (ad < 1.0f) ? 0.5f * d1 * d1 : ad - 0.5f;
            ad = fabsf(d2); sl1 += (ad < 1.0f) ? 0.5f * d2 * d2 : ad - 0.5f;
            ad = fabsf(d3); sl1 += (ad < 1.0f) ? 0.5f * d3 * d3 : ad - 0.5f;
            p_sl1 = sl1;
        }
    }
    // converged here: EXEC all-ones for WMMA
    float r0 = wave_sum_wmma(p_sl1);
    float r1 = wave_sum_wmma(p_ce);
    float r2 = wave_sum_wmma(p_np);
    __shared__ float sw[TPB / 32][3];
    int wid = tid >> 5, lane = tid & 31;
    if (lane == 0) { sw[wid][0] = r0; sw[wid][1] = r1; sw[wid][2] = r2; }
    __syncthreads();
    if (tid == 0) {
        float a0 = 0, a1 = 0, a2 = 0;
        for (int w = 0; w < TPB / 32; ++w) { a0 += sw[w][0]; a1 += sw[w][1]; a2 += sw[w][2]; }
        float* p = part + ((size_t)b * NBLK + blockIdx.x) * 3;
        p[0] = a0; p[1] = a1; p[2] = a2;
    }
}

// Per-batch: combine block partials (fixed order), radix-select the K-th largest
// negative score in LDS, sum top-K negatives (tie-corrected), emit per-batch terms.
__global__ void __launch_bounds__(TPB)
k_hardneg(const float* __restrict__ neg, const float* __restrict__ part,
          float* __restrict__ bsum) {
    int b = blockIdx.x, tid = threadIdx.x;
    __shared__ float sneg[HW];
    __shared__ float sred[TPB];
    __shared__ int   scnt;
    __shared__ float stats[3];
    if (tid == 0) {
        float a0 = 0, a1 = 0, a2 = 0;
        for (int i = 0; i < NBLK; ++i) {
            const float* p = part + ((size_t)b * NBLK + i) * 3;
            a0 += p[0]; a1 += p[1]; a2 += p[2];
        }
        stats[0] = a0; stats[1] = a1; stats[2] = a2;
    }
    for (int i = tid; i < HW; i += TPB) sneg[i] = neg[(size_t)b * HW + i];
    __syncthreads();
    float sum_sl1 = stats[0], sum_ce = stats[1], npf = stats[2];
    int npos = (int)(npf + 0.5f);
    int K = 3 * npos;
    int nneg = HW - npos;
    if (K > nneg) K = nneg;
    float nums = fmaxf(npf, EPS16F);
    if (K > 0) {
        // find K-th largest monotonic key via 32-step binary search
        unsigned lo = 0u, hi = 0xFFFFFFFFu;
        while (lo < hi) {
            unsigned mid = (unsigned)(((unsigned long long)lo + (unsigned long long)hi + 1ull) >> 1);
            if (tid == 0) scnt = 0;
            __syncthreads();
            int lc = 0;
            for (int i = tid; i < HW; i += TPB) {
                unsigned u = __float_as_uint(sneg[i]);
                unsigned k = (u & 0x80000000u) ? ~u : (u | 0x80000000u);
                lc += (k >= mid) ? 1 : 0;
            }
            atomicAdd(&scnt, lc);
            __syncthreads();
            int c = scnt;
            __syncthreads();
            if (c >= K) lo = mid; else hi = mid - 1;
        }
        unsigned T = lo;
        if (tid == 0) scnt = 0;
        __syncthreads();
        float ls = 0.0f; int lc = 0;
        for (int i = tid; i < HW; i += TPB) {
            float f = sneg[i];
            unsigned u = __float_as_uint(f);
            unsigned k = (u & 0x80000000u) ? ~u : (u | 0x80000000u);
            if (k > T) { ls += f; lc += 1; }
        }
        sred[tid] = ls;
        atomicAdd(&scnt, lc);
        __syncthreads();
        if (tid == 0) {
            float tot = 0.0f;
            for (int i = 0; i < TPB; ++i) tot += sred[i];   // deterministic order
            int cgt = scnt;
            unsigned ut = (T & 0x80000000u) ? (T & 0x7FFFFFFFu) : ~T;
            float tv = __uint_as_float(ut);
            float sum_hn = tot + (float)(K - cgt) * tv;     // ties at threshold
            float* o = bsum + (size_t)b * 4;
            o[0] = sum_sl1; o[1] = npf; o[2] = sum_ce / nums; o[3] = sum_hn / nums;
        }
    } else if (tid == 0) {
        float* o = bsum + (size_t)b * 4;
        o[0] = sum_sl1; o[1] = npf; o[2] = sum_ce / nums; o[3] = 0.0f;
    }
}

__global__ void k_final(const float* __restrict__ bsum, float* __restrict__ out) {
    if (threadIdx.x == 0 && blockIdx.x == 0) {
        float tsl1 = 0, tnp = 0, apos = 0, aneg = 0;
        for (int b = 0; b < BATCH; ++b) {
            const float* o = bsum + (size_t)b * 4;
            tsl1 += o[0]; tnp += o[1]; apos += o[2]; aneg += o[3];
        }
        float l_box = tsl1 / fmaxf(tnp, 1.0f);
        out[0] = l_box + apos / (float)BATCH + aneg / (float)BATCH;
    }
}

extern "C" void kernel_launch(void* const* d_in, const int* in_sizes, int n_in,
                              void* d_out, int out_size, void* d_ws, size_t ws_size,
                              hipStream_t stream) {
    (void)in_sizes; (void)n_in; (void)out_size; (void)ws_size;
    const float* preg    = (const float*)d_in[0];
    const float* pcls    = (const float*)d_in[1];
    const float* gboxes  = (const float*)d_in[2];
    const int*   glabels = (const int*)d_in[3];
    const float* ancs    = (const float*)d_in[4];
    float* out = (float*)d_out;

    char* ws = (char*)d_ws;
    size_t off = 0;
    unsigned long long* best = (unsigned long long*)(ws + off); off += (size_t)BATCH * NGT * 8;
    float* max_iou = (float*)(ws + off); off += (size_t)BATCH * HW * 4;
    int*   bidx    = (int*)  (ws + off); off += (size_t)BATCH * HW * 4;
    float* neg     = (float*)(ws + off); off += (size_t)BATCH * HW * 4;
    float* part    = (float*)(ws + off); off += (size_t)BATCH * NBLK * 3 * 4;
    float* bsum    = (float*)(ws + off); off += (size_t)BATCH * 4 * 4;

    k_init   <<<dim3((BATCH * NGT + 255) / 256), dim3(256), 0, stream>>>(best);
    k_iou    <<<dim3(NBLK, BATCH), dim3(TPB), 0, stream>>>(gboxes, ancs, max_iou, bidx, best);
    k_force  <<<dim3(1), dim3(BATCH), 0, stream>>>(best, max_iou, bidx);
    k_main   <<<dim3(NBLK, BATCH), dim3(TPB), 0, stream>>>(preg, pcls, gboxes, glabels, ancs,
                                                           max_iou, bidx, neg, part);
    k_hardneg<<<dim3(BATCH), dim3(TPB), 0, stream>>>(neg, part, bsum);
    k_final  <<<dim3(1), dim3(1), 0, stream>>>(bsum, out);
}